// SNSMotionVisionEye_44796508897727
// MI455X (gfx1250) — compile-verified
//
#include <hip/hip_runtime.h>

// ---------------------------------------------------------------------------
// SNS motion-vision eye, fused for MI455X (gfx1250, wave32, WMMA).
//   B=16, H=W=512, P=508 (5x5 valid), EM=506 (3x3 valid)
// Conv MACs go to the matrix pipe via V_WMMA_F32_16X16X4_F32; everything else
// is a single elementwise pass so the kernel sits on the HBM roofline.
// ---------------------------------------------------------------------------

#define NB   16
#define HH   512
#define PP   508
#define EMD  506
#define TR   16      // output rows per block tile
#define TC   128     // output cols per block tile
#define LDA  132     // LDS row stride (TC + 4 halo)

typedef float v2f __attribute__((ext_vector_type(2)));
typedef float v8f __attribute__((ext_vector_type(8)));

__device__ __forceinline__ float clip01(float v) {
    return __builtin_fminf(__builtin_fmaxf(v, 0.0f), 1.0f);
}

// Output layout (flat concat, return order):
//   [0]              n_input        16*512*512
//   [1..15]         P-domain outs   16*508*508 each
//   [16]            n_emd           16*506*506
static constexpr size_t S512 = (size_t)NB * HH * HH;     // 4,194,304
static constexpr size_t SP   = (size_t)NB * PP * PP;     // 4,129,024

__global__ __launch_bounds__(256)
void sns_fused_kernel(const float* __restrict__ state_input,
                      const float* __restrict__ bp_on_input,
                      const float* __restrict__ bp_on_fast,
                      const float* __restrict__ bp_on_slow,
                      const float* __restrict__ bp_on_out,
                      const float* __restrict__ state_lowpass,
                      const float* __restrict__ bp_off_input,
                      const float* __restrict__ bp_off_fast,
                      const float* __restrict__ bp_off_slow,
                      const float* __restrict__ bp_off_out,
                      const float* __restrict__ direct_on,
                      const float* __restrict__ enhance_on,
                      const float* __restrict__ suppress_on,
                      const float* __restrict__ direct_off,
                      const float* __restrict__ enhance_off,
                      const float* __restrict__ suppress_off,
                      const float* __restrict__ emd_cw_on,
                      const float* __restrict__ taus,
                      const float* __restrict__ g_elem,
                      const float* __restrict__ E_elem,
                      const float* __restrict__ g_conv5,
                      const float* __restrict__ E_conv5,
                      const float* __restrict__ g_conv3,
                      const float* __restrict__ E_conv3,
                      float* __restrict__ dout) {
    __shared__ float sA[20 * LDA];        // clip(state_input), +4 halo
    __shared__ float sT[3 * 18 * LDA];    // raw enhance_on / direct_on / suppress_on, +2 halo
    __shared__ float sW5[28 * 16];        // 5x5 WMMA A-operand: [tap k][filter m]
    __shared__ float sW3[28 * 16];        // 3x3 WMMA A-operand (27 taps across 3 tensors)
    __shared__ float sPar[51];            // taus | g_elem | E_elem

    const int b   = blockIdx.z;
    const int r0  = blockIdx.y * TR;
    const int c0  = blockIdx.x * TC;
    const int tid = threadIdx.x;

    // ---- cooperative staging ------------------------------------------------
    for (int i = tid; i < 51; i += 256)
        sPar[i] = (i < 17) ? taus[i] : (i < 34) ? g_elem[i - 17] : E_elem[i - 34];

    for (int i = tid; i < 28 * 16; i += 256) {
        int k = i >> 4, m = i & 15;
        float w = 0.0f;
        if (k < 25) {
            if (m < 3)      w = g_conv5[m * 25 + k] * E_conv5[m * 25 + k];
            else if (m < 6) w = g_conv5[(m - 3) * 25 + k];
        }
        sW5[i] = w;
        float w3 = 0.0f;
        if (k < 27) {
            int t = k / 9, j = k % 9;
            if (m == 0)      w3 = g_conv3[t * 9 + j] * E_conv3[t * 9 + j];
            else if (m == 1) w3 = g_conv3[t * 9 + j];
        }
        sW3[i] = w3;
    }

    for (int i = tid; i < 20 * LDA; i += 256) {
        int r = i / LDA, c = i % LDA;
        int gr = r0 + r, gc = c0 + c;
        float v = 0.0f;
        if (gr < HH && gc < HH)
            v = clip01(state_input[((size_t)b * HH + gr) * HH + gc]);
        sA[i] = v;
    }
    for (int i = tid; i < 18 * LDA; i += 256) {
        int r = i / LDA, c = i % LDA;
        int gr = r0 + r, gc = c0 + c;
        bool ok = (gr < PP) && (gc < PP);
        size_t gi = ((size_t)b * PP + gr) * PP + gc;
        sT[0 * 18 * LDA + i] = ok ? enhance_on[gi]  : 0.0f;
        sT[1 * 18 * LDA + i] = ok ? direct_on[gi]   : 0.0f;
        sT[2 * 18 * LDA + i] = ok ? suppress_on[gi] : 0.0f;
    }
    __syncthreads();

    // ---- per-wave WMMA operand setup ---------------------------------------
    const int wv   = tid >> 5;
    const int lane = tid & 31;
    const int ln   = lane & 15;   // M (filter) for A, N (pixel) for B, col in D
    const int kh   = lane >> 4;   // K half: lanes 0-15 hold K={4q,4q+1}, 16-31 hold {4q+2,4q+3}

    v2f a5[7], a3[7];
    int o5x[7], o5y[7], o3x[7], o3y[7];
#pragma unroll
    for (int q = 0; q < 7; ++q) {
        int k0 = 4 * q + 2 * kh;
        a5[q].x = sW5[k0 * 16 + ln];
        a5[q].y = sW5[(k0 + 1) * 16 + ln];
        a3[q].x = sW3[k0 * 16 + ln];
        a3[q].y = sW3[(k0 + 1) * 16 + ln];
        int kk0 = (k0     < 25) ? k0     : 24;   // padded taps: weight==0, addr clamped
        int kk1 = (k0 + 1 < 25) ? k0 + 1 : 24;
        o5x[q] = (kk0 / 5) * LDA + (kk0 % 5);
        o5y[q] = (kk1 / 5) * LDA + (kk1 % 5);
        int m0 = (k0     < 27) ? k0     : 26;
        int m1 = (k0 + 1 < 27) ? k0 + 1 : 26;
        o3x[q] = (m0 / 9) * (18 * LDA) + ((m0 % 9) / 3) * LDA + (m0 % 9) % 3;
        o3y[q] = (m1 / 9) * (18 * LDA) + ((m1 % 9) / 3) * LDA + (m1 % 9) % 3;
    }

    const float* tau = sPar;
    const float* ge  = sPar + 17;
    const float* Ee  = sPar + 34;

    const int rA = 2 * wv;       // each wave owns a row pair of the tile
    const int rB = 2 * wv + 1;

    auto chain5 = [&](int rl, int coln) -> v8f {
        v8f acc = {};
        int base = rl * LDA + coln;
#pragma unroll
        for (int q = 0; q < 7; ++q) {
            v2f bb;
            bb.x = sA[base + o5x[q]];
            bb.y = sA[base + o5y[q]];
            acc = __builtin_amdgcn_wmma_f32_16x16x4_f32(
                false, a5[q], false, bb, (short)0, acc, false, false);
        }
        return acc;
    };
    auto chain3 = [&](int rl, int coln) -> v8f {
        v8f acc = {};
        int base = rl * LDA + coln;
#pragma unroll
        for (int q = 0; q < 7; ++q) {
            v2f bb;
            bb.x = clip01(sT[base + o3x[q]]);
            bb.y = clip01(sT[base + o3y[q]]);
            acc = __builtin_amdgcn_wmma_f32_16x16x4_f32(
                false, a3[q], false, bb, (short)0, acc, false, false);
        }
        return acc;
    };

    // ---- main loop: 8 strips of 16 pixels per row pair ----------------------
    for (int s = 0; s < 8; ++s) {
        const int cb   = 16 * s;
        const int coln = cb + ln;

        v8f d5A = chain5(rA, coln);
        v8f d5B = chain5(rB, coln);
        v8f d3A = chain3(rA, coln);
        v8f d3B = chain3(rB, coln);

        // lanes 0-15 -> row rA, lanes 16-31 -> row rB (shuffle row B down)
        float c5[6];
#pragma unroll
        for (int j = 0; j < 6; ++j) {
            float hi = __shfl(d5B[j], ln, 32);
            c5[j] = (lane < 16) ? d5A[j] : hi;
        }
        float c3gE, c3g;
        { float hi = __shfl(d3B[0], ln, 32); c3gE = (lane < 16) ? d3A[0] : hi; }
        { float hi = __shfl(d3B[1], ln, 32); c3g  = (lane < 16) ? d3A[1] : hi; }

        const int rl = (lane < 16) ? rA : rB;
        const int gy = r0 + rl;
        const int gx = c0 + coln;
        const bool valid = (gy < PP) && (gx < PP);
        const size_t ip = ((size_t)b * PP + gy) * PP + gx;

        float v_bpi = valid ? bp_on_input[ip]   : 0.0f;
        float v_bpf = valid ? bp_on_fast[ip]    : 0.0f;
        float v_bps = valid ? bp_on_slow[ip]    : 0.0f;
        float v_bpo = valid ? bp_on_out[ip]     : 0.0f;
        float v_lp  = valid ? state_lowpass[ip] : 0.0f;
        float v_ofi = valid ? bp_off_input[ip]  : 0.0f;
        float v_off = valid ? bp_off_fast[ip]   : 0.0f;
        float v_ofs = valid ? bp_off_slow[ip]   : 0.0f;
        float v_ofo = valid ? bp_off_out[ip]    : 0.0f;
        float v_dof = valid ? direct_off[ip]    : 0.0f;
        float v_eof = valid ? enhance_off[ip]   : 0.0f;
        float v_sof = valid ? suppress_off[ip]  : 0.0f;
        const int loff = rl * LDA + coln;
        float v_eon = sT[0 * 18 * LDA + loff];
        float v_don = sT[1 * 18 * LDA + loff];
        float v_son = sT[2 * 18 * LDA + loff];

        // conv-synapse currents: I = conv(a, g*E) - post * conv(a, g)
        float in_bpon  = c5[0] - v_bpi * c5[3];
        float in_lp    = c5[1] - v_lp  * c5[4];
        float in_bpoff = c5[2] - v_ofi * c5[5];

        // ON bandpass (g/E 0..3, taus 1..4)
        float a_bpi = clip01(v_bpi), a_bpf = clip01(v_bpf), a_bps = clip01(v_bps);
        float i2f = ge[0] * a_bpi * (Ee[0] - v_bpf);
        float i2s = ge[1] * a_bpi * (Ee[1] - v_bps);
        float f2o = ge[2] * a_bpf * (Ee[2] - v_bpo);
        float s2o = ge[3] * a_bps * (Ee[3] - v_bpo);
        float o1 = v_bpi + tau[1] * (in_bpon - v_bpi);
        float o2 = v_bpf + tau[2] * (i2f - v_bpf);
        float o3 = v_bps + tau[3] * (i2s - v_bps);
        float o4 = v_bpo + tau[4] * ((f2o + s2o) - v_bpo);
        float o5 = v_lp  + tau[5] * (in_lp - v_lp);

        // OFF bandpass (g/E 4..7, taus 6..9)
        float a_ofi = clip01(v_ofi), a_of2 = clip01(v_off), a_os2 = clip01(v_ofs);
        float q2f = ge[4] * a_ofi * (Ee[4] - v_off);
        float q2s = ge[5] * a_ofi * (Ee[5] - v_ofs);
        float g2o = ge[6] * a_of2 * (Ee[6] - v_ofo);
        float h2o = ge[7] * a_os2 * (Ee[7] - v_ofo);
        float o6 = v_ofi + tau[6] * (in_bpoff - v_ofi);
        float o7 = v_off + tau[7] * (q2f - v_off);
        float o8 = v_ofs + tau[8] * (q2s - v_ofs);
        float o9 = v_ofo + tau[9] * ((g2o + h2o) - v_ofo);

        // lateral circuits
        float a_bpo = clip01(v_bpo), a_lp = clip01(v_lp), a_don = clip01(v_don);
        float a_ofo = clip01(v_ofo), a_sof = clip01(v_sof);
        float a_dof = clip01(v_dof), a_eof = clip01(v_eof);
        float dir_on_in  = ge[8]  * a_bpo * (Ee[8]  - v_don);
        float enh_on_in  = ge[9]  * a_lp  * (Ee[9]  - v_eon);
        float sup_on_in  = ge[10] * a_don * (Ee[10] - v_son);
        float dir_off_in = ge[11] * a_ofo * (Ee[11] - v_dof)
                         + ge[12] * a_sof * (Ee[12] - v_dof);
        float enh_off_in = ge[13] * a_lp  * (Ee[13] - v_eof)
                         + ge[14] * a_sof * (Ee[14] - v_eof);
        float sup_off_in = ge[15] * a_dof * (Ee[15] - v_sof)
                         + ge[16] * a_eof * (Ee[16] - v_sof);
        float o10 = v_don + tau[10] * (dir_on_in  - v_don);
        float o11 = v_eon + tau[11] * (enh_on_in  - v_eon);
        float o12 = v_son + tau[12] * (sup_on_in  - v_son);
        float o13 = v_dof + tau[13] * (dir_off_in - v_dof);
        float o14 = v_eof + tau[14] * (enh_off_in - v_eof);
        float o15 = v_sof + tau[15] * (sup_off_in - v_sof);

        if (valid) {
            float* po = dout + S512;
            float ov[15] = {o1, o2, o3, o4, o5, o6, o7, o8,
                            o9, o10, o11, o12, o13, o14, o15};
#pragma unroll
            for (int j = 0; j < 15; ++j)
                po[(size_t)j * SP + ip] = ov[j];
        }

        if (gy < EMD && gx < EMD) {
            size_t ie = ((size_t)b * EMD + gy) * EMD + gx;
            float post   = emd_cw_on[ie];
            float emd_in = c3gE - post * c3g;
            dout[S512 + 15 * SP + ie] = post + tau[16] * (emd_in - post);
        }
    }
}

// n_input = integ(state_input, x, taus[0]) over the full 512x512 grid.
__global__ __launch_bounds__(256)
void sns_input_kernel(const float* __restrict__ x,
                      const float* __restrict__ s,
                      const float* __restrict__ taus,
                      float* __restrict__ dout) {
    size_t i = (size_t)blockIdx.x * blockDim.x + threadIdx.x;
    size_t n4 = S512 / 4;
    if (i < n4) {
        float t0 = taus[0];
        float4 xv = ((const float4*)x)[i];
        float4 sv = ((const float4*)s)[i];
        float4 r;
        r.x = sv.x + t0 * (xv.x - sv.x);
        r.y = sv.y + t0 * (xv.y - sv.y);
        r.z = sv.z + t0 * (xv.z - sv.z);
        r.w = sv.w + t0 * (xv.w - sv.w);
        ((float4*)dout)[i] = r;
    }
}

extern "C" void kernel_launch(void* const* d_in, const int* in_sizes, int n_in,
                              void* d_out, int out_size, void* d_ws, size_t ws_size,
                              hipStream_t stream) {
    (void)in_sizes; (void)n_in; (void)out_size; (void)d_ws; (void)ws_size;
    const float* x_in         = (const float*)d_in[0];
    const float* state_input  = (const float*)d_in[1];
    const float* bp_on_input  = (const float*)d_in[2];
    const float* bp_on_fast   = (const float*)d_in[3];
    const float* bp_on_slow   = (const float*)d_in[4];
    const float* bp_on_out    = (const float*)d_in[5];
    const float* state_lowpss = (const float*)d_in[6];
    const float* bp_off_input = (const float*)d_in[7];
    const float* bp_off_fast  = (const float*)d_in[8];
    const float* bp_off_slow  = (const float*)d_in[9];
    const float* bp_off_out   = (const float*)d_in[10];
    const float* direct_on    = (const float*)d_in[11];
    const float* enhance_on   = (const float*)d_in[12];
    const float* suppress_on  = (const float*)d_in[13];
    const float* direct_off   = (const float*)d_in[14];
    const float* enhance_off  = (const float*)d_in[15];
    const float* suppress_off = (const float*)d_in[16];
    const float* emd_cw_on    = (const float*)d_in[17];
    const float* taus         = (const float*)d_in[18];
    const float* g_elem       = (const float*)d_in[19];
    const float* E_elem       = (const float*)d_in[20];
    const float* g_conv5      = (const float*)d_in[21];
    const float* E_conv5      = (const float*)d_in[22];
    const float* g_conv3      = (const float*)d_in[23];
    const float* E_conv3      = (const float*)d_in[24];
    float* out = (float*)d_out;

    dim3 grid((PP + TC - 1) / TC, (PP + TR - 1) / TR, NB);   // 4 x 32 x 16
    dim3 block(256);
    sns_fused_kernel<<<grid, block, 0, stream>>>(
        state_input, bp_on_input, bp_on_fast, bp_on_slow, bp_on_out,
        state_lowpss, bp_off_input, bp_off_fast, bp_off_slow, bp_off_out,
        direct_on, enhance_on, suppress_on, direct_off, enhance_off,
        suppress_off, emd_cw_on, taus, g_elem, E_elem,
        g_conv5, E_conv5, g_conv3, E_conv3, out);

    size_t n4 = S512 / 4;
    dim3 grid2((unsigned)((n4 + 255) / 256));
    sns_input_kernel<<<grid2, block, 0, stream>>>(x_in, state_input, taus, out);
}